// DAModule_4466765988504
// MI455X (gfx1250) — compile-verified
//
#include <hip/hip_runtime.h>
#include <math.h>

typedef __attribute__((ext_vector_type(2))) float v2f;
typedef __attribute__((ext_vector_type(8))) float v8f;

__device__ __forceinline__ v8f wmma4(v2f a, v2f b, v8f c) {
  // V_WMMA_F32_16X16X4_F32 : D = A(16x4,f32) * B(4x16,f32) + C(16x16,f32)
  return __builtin_amdgcn_wmma_f32_16x16x4_f32(false, a, false, b, (short)0, c,
                                               false, false);
}

__device__ __forceinline__ float hswish(float x) {
  float t = fminf(fmaxf(x + 3.0f, 0.0f), 6.0f);
  return x * t * (1.0f / 6.0f);
}

// Low 32 bits of a flat shared-aperture address = LDS byte offset.
__device__ __forceinline__ unsigned lds_off(const void* p) {
  return (unsigned)(uintptr_t)p;
}

// CDNA5 async DMA: global -> LDS, 16B per lane, tracked by ASYNCcnt.
__device__ __forceinline__ void async_to_lds_b128(unsigned ldsoff,
                                                  const float* g) {
  asm volatile("global_load_async_to_lds_b128 %0, %1, off"
               :: "v"(ldsoff), "v"(g)
               : "memory");
}
__device__ __forceinline__ void wait_asynccnt0() {
  asm volatile("s_wait_asynccnt 0x0" ::: "memory");
}

// ---------------------------------------------------------------------------
// conv1x1 as GEMM: Y[b][m][n] = sum_k W[m][k] * X[b][k][n], K fixed = 256.
// block = (32,4): 4 waves share one A tile (staged to LDS by async DMA),
// each wave owns one 16x16 output tile.
// ---------------------------------------------------------------------------
#define ALD 260  // padded LDS row stride (floats): bank = lane*4, conflict-free
template <int HS>
__global__ __launch_bounds__(128)
void gemm_wmma_kernel(const float* __restrict__ W, const float* __restrict__ X,
                      float* __restrict__ Y, int M, int N) {
  __shared__ float alds[16 * ALD];
  const int tid = threadIdx.y * 32 + threadIdx.x;
  const int lane = threadIdx.x & 31;
  const int half = lane >> 4, lo = lane & 15;
  const int n0 = (blockIdx.x * blockDim.y + threadIdx.y) * 16;
  const int m0 = blockIdx.y * 16;
  const float* Xb = X + (size_t)blockIdx.z * 256 * N;
  float* Yb = Y + (size_t)blockIdx.z * M * N;

  // Stage A tile (16 rows x 256) into LDS: 1024 x B128 DMA transfers.
  const unsigned abase = lds_off(alds);
  for (int i = tid; i < 1024; i += 128) {
    int row = i >> 6, c4 = (i & 63) << 2;
    async_to_lds_b128(abase + (unsigned)(row * ALD + c4) * 4u,
                      W + (size_t)(m0 + row) * 256 + c4);
  }
  wait_asynccnt0();
  __syncthreads();

  const int kr = 2 * half;
  const float* ap = alds + lo * ALD + kr;
  const float* xp = Xb + (size_t)kr * N + n0 + lo;
  v8f acc = {0.f, 0.f, 0.f, 0.f, 0.f, 0.f, 0.f, 0.f};
#pragma unroll 4
  for (int k0 = 0; k0 < 256; k0 += 4) {
    v2f a, b;
    a.x = ap[0];
    a.y = ap[1];
    float b0 = xp[0];
    float b1 = xp[N];
    if (HS) { b0 = hswish(b0); b1 = hswish(b1); }
    b.x = b0; b.y = b1;
    acc = wmma4(a, b, acc);
    ap += 4;
    xp += (size_t)4 * N;
  }
#pragma unroll
  for (int r = 0; r < 8; ++r)
    Yb[(size_t)(m0 + r + 8 * half) * N + n0 + lo] = acc[r];
}

// ---------------------------------------------------------------------------
// In-place GroupNorm over [b][C][N]; one block per (b, group).
// ---------------------------------------------------------------------------
__global__ __launch_bounds__(256)
void gn_kernel(float* __restrict__ Y, const float* __restrict__ gamma,
               const float* __restrict__ beta, int C, int N, int groups,
               float eps) {
  __shared__ float red[512];
  const int grp = blockIdx.x % groups;
  const int b = blockIdx.x / groups;
  const int cpg = C / groups;
  const int cnt = cpg * N;
  float* base = Y + ((size_t)b * C + (size_t)grp * cpg) * N;
  float s = 0.f, q = 0.f;
  for (int i = threadIdx.x; i < cnt; i += 256) {
    float v = base[i]; s += v; q += v * v;
  }
  red[threadIdx.x] = s; red[256 + threadIdx.x] = q;
  __syncthreads();
  for (int off = 128; off > 0; off >>= 1) {
    if (threadIdx.x < off) {
      red[threadIdx.x] += red[threadIdx.x + off];
      red[256 + threadIdx.x] += red[256 + threadIdx.x + off];
    }
    __syncthreads();
  }
  const float mu = red[0] / (float)cnt;
  const float var = red[256] / (float)cnt - mu * mu;
  const float rstd = rsqrtf(var + eps);
  for (int i = threadIdx.x; i < cnt; i += 256) {
    int c = i / N;
    base[i] = (base[i] - mu) * rstd * gamma[grp * cpg + c] + beta[grp * cpg + c];
  }
}

// ---------------------------------------------------------------------------
// Offset branch part 1: depthwise 3x3 stride-2 pad-1 conv on q-groups.
// ---------------------------------------------------------------------------
__global__ __launch_bounds__(256)
void offset_conv_kernel(const float* __restrict__ q,
                        const float* __restrict__ w_dw,
                        float* __restrict__ convb) {
  const int n = blockIdx.x, b = n >> 2, g = n & 3;
  const float* qb = q + ((size_t)b * 256 + g * 64) * 1024;
  for (int idx = threadIdx.x; idx < 64 * 256; idx += 256) {
    int c = idx >> 8, s = idx & 255;
    int oy = s >> 4, ox = s & 15;
    float acc = 0.f;
#pragma unroll
    for (int ky = 0; ky < 3; ++ky) {
      int y = 2 * oy - 1 + ky;
      if (y < 0 || y >= 32) continue;
#pragma unroll
      for (int kx = 0; kx < 3; ++kx) {
        int x = 2 * ox - 1 + kx;
        if (x < 0 || x >= 32) continue;
        acc += w_dw[c * 9 + ky * 3 + kx] * qb[(size_t)c * 1024 + y * 32 + x];
      }
    }
    convb[(size_t)n * 16384 + idx] = acc;
  }
}

// ---------------------------------------------------------------------------
// Offset branch part 2: GN(2) -> hardswish -> 1x1 to 2ch -> +ref -> clamp.
// ---------------------------------------------------------------------------
__global__ __launch_bounds__(256)
void offset_pos_kernel(const float* __restrict__ convb,
                       const float* __restrict__ g_off,
                       const float* __restrict__ b_off,
                       const float* __restrict__ w_off,
                       float* __restrict__ pos) {
  __shared__ float red[1024];
  __shared__ float stats[4];
  const int n = blockIdx.x;
  const int tid = threadIdx.x;
  const float* cb = convb + (size_t)n * 16384;
  float s0 = 0, q0 = 0, s1 = 0, q1 = 0;
  for (int idx = tid; idx < 16384; idx += 256) {
    float v = cb[idx];
    if ((idx >> 8) < 32) { s0 += v; q0 += v * v; } else { s1 += v; q1 += v * v; }
  }
  red[tid] = s0; red[256 + tid] = q0; red[512 + tid] = s1; red[768 + tid] = q1;
  __syncthreads();
  for (int off = 128; off > 0; off >>= 1) {
    if (tid < off) {
      red[tid] += red[tid + off];
      red[256 + tid] += red[256 + tid + off];
      red[512 + tid] += red[512 + tid + off];
      red[768 + tid] += red[768 + tid + off];
    }
    __syncthreads();
  }
  if (tid == 0) {
    const float cnt = 32.f * 256.f;
    float mu0 = red[0] / cnt, var0 = red[256] / cnt - mu0 * mu0;
    float mu1 = red[512] / cnt, var1 = red[768] / cnt - mu1 * mu1;
    stats[0] = mu0; stats[1] = rsqrtf(var0 + 1e-5f);
    stats[2] = mu1; stats[3] = rsqrtf(var1 + 1e-5f);
  }
  __syncthreads();
  const int s = tid, oy = s >> 4, ox = s & 15;
  float ay = 0.f, ax = 0.f;
  for (int c = 0; c < 64; ++c) {
    float mu = stats[(c >> 5) * 2], rstd = stats[(c >> 5) * 2 + 1];
    float v = hswish((cb[c * 256 + s] - mu) * rstd * g_off[c] + b_off[c]);
    ay += w_off[c] * v;
    ax += w_off[64 + c] * v;
  }
  float ry = ((0.5f + (float)oy) * (1.0f / 15.0f)) * 2.0f - 1.0f;
  float rx = ((0.5f + (float)ox) * (1.0f / 15.0f)) * 2.0f - 1.0f;
  pos[((size_t)n * 256 + s) * 2 + 0] = fminf(fmaxf(ay + ry, -1.f), 1.f);
  pos[((size_t)n * 256 + s) * 2 + 1] = fminf(fmaxf(ax + rx, -1.f), 1.f);
}

// ---------------------------------------------------------------------------
// Bilinear grid-sample of key/value at pos -> k_s/v_s [B][256][256].
// ---------------------------------------------------------------------------
__global__ __launch_bounds__(256)
void sample_kernel(const float* __restrict__ key, const float* __restrict__ value,
                   const float* __restrict__ pos, float* __restrict__ ks,
                   float* __restrict__ vs) {
  const int n = blockIdx.x, b = n >> 2, g = n & 3;
  const int s = threadIdx.x;
  float py = pos[((size_t)n * 256 + s) * 2 + 0];
  float px = pos[((size_t)n * 256 + s) * 2 + 1];
  float x = (px + 1.f) * 0.5f * 31.f;
  float y = (py + 1.f) * 0.5f * 31.f;
  float xf = floorf(x), yf = floorf(y);
  float wx = x - xf, wy = y - yf;
  int x0 = min(max((int)xf, 0), 31), x1 = min(x0 + 1, 31);
  int y0 = min(max((int)yf, 0), 31), y1 = min(y0 + 1, 31);
  float w00 = (1 - wx) * (1 - wy), w01 = wx * (1 - wy);
  float w10 = (1 - wx) * wy, w11 = wx * wy;
  for (int cc = 0; cc < 64; ++cc) {
    size_t ib = ((size_t)b * 256 + g * 64 + cc) * 1024;
    size_t ob = ((size_t)b * 256 + g * 64 + cc) * 256 + s;
    ks[ob] = key[ib + y0 * 32 + x0] * w00 + key[ib + y0 * 32 + x1] * w01 +
             key[ib + y1 * 32 + x0] * w10 + key[ib + y1 * 32 + x1] * w11;
    vs[ob] = value[ib + y0 * 32 + x0] * w00 + value[ib + y0 * 32 + x1] * w01 +
             value[ib + y1 * 32 + x0] * w10 + value[ib + y1 * 32 + x1] * w11;
  }
}

// ---------------------------------------------------------------------------
// Fused flash-style attention: per (head bh, 16-query tile m0):
//   S = (qh^T kh)*scale + bilinear(rpe) ; P = softmax(S) ; ao = P @ vh^T
// kh/vh tiles DMA'd to LDS (async B128); rpe + S/P live in LDS too.
// ---------------------------------------------------------------------------
#define SLD 260
__global__ __launch_bounds__(256)
void attn_kernel(const float* __restrict__ q, const float* __restrict__ kh,
                 const float* __restrict__ vh, const float* __restrict__ pos,
                 const float* __restrict__ rpe, float* __restrict__ ao) {
  __shared__ float klds[32 * SLD];
  __shared__ float vlds[32 * SLD];
  __shared__ float rpe_lds[63 * 63];
  __shared__ float S[16 * SLD];
  __shared__ float posy[256], posx[256];
  __shared__ float red[256];
  __shared__ float rowstat[16];
  const int tid = threadIdx.x;
  const int bh = blockIdx.y;
  const int b = bh >> 3, h = bh & 7;
  const int bg = b * 4 + (h >> 1);
  const int m0 = blockIdx.x * 16;
  const size_t qbase = ((size_t)b * 256 + h * 32) * 1024;
  const size_t kbase = ((size_t)b * 256 + h * 32) * 256;

  // Async DMA: kh/vh head tiles (32 ch x 256 n) -> padded LDS.
  const unsigned kb = lds_off(klds), vb = lds_off(vlds);
  for (int i = tid; i < 2048; i += 256) {
    int row = i >> 6, c4 = (i & 63) << 2;
    unsigned loff = (unsigned)(row * SLD + c4) * 4u;
    async_to_lds_b128(kb + loff, kh + kbase + row * 256 + c4);
    async_to_lds_b128(vb + loff, vh + kbase + row * 256 + c4);
  }
  for (int i = tid; i < 63 * 63; i += 256) rpe_lds[i] = rpe[(size_t)h * 3969 + i];
  posy[tid] = pos[((size_t)bg * 256 + tid) * 2 + 0];
  posx[tid] = pos[((size_t)bg * 256 + tid) * 2 + 1];
  wait_asynccnt0();
  __syncthreads();

  const int wid = tid >> 5, lane = tid & 31, half = lane >> 4, lo = lane & 15;
  const int kr = 2 * half;
  const float scale = 0.17677669529663687f;  // 32^-0.5

  // ---- S = QK^T * scale + rpe bias ; each wave: 2 n-tiles of 16 ----
  for (int sub = 0; sub < 2; ++sub) {
    const int n0 = (wid * 2 + sub) * 16;
    v8f acc = {0.f, 0.f, 0.f, 0.f, 0.f, 0.f, 0.f, 0.f};
#pragma unroll 4
    for (int c0 = 0; c0 < 32; c0 += 4) {
      v2f a, bv;
      a.x = q[qbase + (size_t)(c0 + kr) * 1024 + m0 + lo];
      a.y = q[qbase + (size_t)(c0 + kr + 1) * 1024 + m0 + lo];
      bv.x = klds[(c0 + kr) * SLD + n0 + lo];
      bv.y = klds[(c0 + kr + 1) * SLD + n0 + lo];
      acc = wmma4(a, bv, acc);
    }
    const int nn = n0 + lo;
    const float pyv = posy[nn], pxv = posx[nn];
#pragma unroll
    for (int r = 0; r < 8; ++r) {
      int ml = r + 8 * half;
      int m = m0 + ml;
      float gy = (float)(m >> 5) * (2.0f / 31.0f) - 1.0f;
      float gx = (float)(m & 31) * (2.0f / 31.0f) - 1.0f;
      float dx = (gx - pxv) * 0.5f;
      float dy = (gy - pyv) * 0.5f;
      float xg = (dx + 1.f) * 0.5f * 62.f;
      float yg = (dy + 1.f) * 0.5f * 62.f;
      float xf = floorf(xg), yf = floorf(yg);
      float wx = xg - xf, wy = yg - yf;
      int x0 = min(max((int)xf, 0), 62), x1 = min(x0 + 1, 62);
      int y0 = min(max((int)yf, 0), 62), y1 = min(y0 + 1, 62);
      float bias = rpe_lds[y0 * 63 + x0] * (1 - wx) * (1 - wy) +
                   rpe_lds[y0 * 63 + x1] * wx * (1 - wy) +
                   rpe_lds[y1 * 63 + x0] * (1 - wx) * wy +
                   rpe_lds[y1 * 63 + x1] * wx * wy;
      S[ml * SLD + nn] = acc[r] * scale + bias;
    }
  }
  __syncthreads();

  // ---- softmax over n=256 per row; 16 threads per row ----
  {
    int row = tid >> 4, j = tid & 15;
    float mx = -1e30f;
    for (int nn = j; nn < 256; nn += 16) mx = fmaxf(mx, S[row * SLD + nn]);
    red[tid] = mx;
    __syncthreads();
    if (j == 0) {
      float m2 = red[tid];
      for (int t = 1; t < 16; ++t) m2 = fmaxf(m2, red[tid + t]);
      rowstat[row] = m2;
    }
    __syncthreads();
    float rmax = rowstat[row];
    float sm = 0.f;
    for (int nn = j; nn < 256; nn += 16) {
      float e = expf(S[row * SLD + nn] - rmax);
      S[row * SLD + nn] = e;
      sm += e;
    }
    red[tid] = sm;
    __syncthreads();
    if (j == 0) {
      float t2 = 0.f;
      for (int t = 0; t < 16; ++t) t2 += red[tid + t];
      rowstat[row] = t2;
    }
    __syncthreads();
    float inv = 1.0f / rowstat[row];
    for (int nn = j; nn < 256; nn += 16) S[row * SLD + nn] *= inv;
  }
  __syncthreads();

  // ---- ao[:, m] = P @ vh^T ; waves 0/1 own the two 16-channel tiles ----
  if (wid < 2) {
    const int ch0 = wid * 16;
    v8f acc = {0.f, 0.f, 0.f, 0.f, 0.f, 0.f, 0.f, 0.f};
#pragma unroll 4
    for (int k0 = 0; k0 < 256; k0 += 4) {
      v2f a, bv;
      a.x = S[lo * SLD + k0 + kr];
      a.y = S[lo * SLD + k0 + kr + 1];
      bv.x = vlds[(ch0 + lo) * SLD + k0 + kr];
      bv.y = vlds[(ch0 + lo) * SLD + k0 + kr + 1];
      acc = wmma4(a, bv, acc);
    }
#pragma unroll
    for (int r = 0; r < 8; ++r)
      ao[qbase + (size_t)(ch0 + lo) * 1024 + m0 + r + 8 * half] = acc[r];
  }
}

// ---------------------------------------------------------------------------
extern "C" void kernel_launch(void* const* d_in, const int* in_sizes, int n_in,
                              void* d_out, int out_size, void* d_ws,
                              size_t ws_size, hipStream_t stream) {
  (void)in_sizes; (void)n_in; (void)out_size; (void)ws_size;
  const float* query = (const float*)d_in[0];
  const float* key   = (const float*)d_in[1];
  const float* value = (const float*)d_in[2];
  const float* w_q   = (const float*)d_in[3];
  const float* gq    = (const float*)d_in[4];
  const float* bq    = (const float*)d_in[5];
  const float* w_k   = (const float*)d_in[6];
  const float* gk    = (const float*)d_in[7];
  const float* bk    = (const float*)d_in[8];
  const float* w_v   = (const float*)d_in[9];
  const float* gv    = (const float*)d_in[10];
  const float* bv    = (const float*)d_in[11];
  const float* w_out = (const float*)d_in[12];
  const float* go    = (const float*)d_in[13];
  const float* bo    = (const float*)d_in[14];
  const float* w_dw  = (const float*)d_in[15];
  const float* g_off = (const float*)d_in[16];
  const float* b_off = (const float*)d_in[17];
  const float* w_off = (const float*)d_in[18];
  const float* rpe   = (const float*)d_in[19];
  float* out = (float*)d_out;

  float* ws   = (float*)d_ws;
  float* qn   = ws;                        // 16*256*1024
  float* posb = qn + 16 * 256 * 1024;      // 64*256*2
  float* ksb  = posb + 64 * 256 * 2;       // 16*256*256
  float* vsb  = ksb + 16 * 256 * 256;      // 16*256*256
  float* khb  = vsb + 16 * 256 * 256;      // 16*256*256
  float* vhb  = khb + 16 * 256 * 256;      // 16*256*256
  float* aob  = vhb + 16 * 256 * 256;      // 16*256*1024
  float* cvb  = aob + 16 * 256 * 1024;     // 64*64*256 (offset dw-conv)

  const dim3 gblk(32, 4, 1);
  // q = GN32(conv1x1(query, w_q))
  gemm_wmma_kernel<0><<<dim3(16, 16, 16), gblk, 0, stream>>>(w_q, query, qn, 256, 1024);
  gn_kernel<<<16 * 32, 256, 0, stream>>>(qn, gq, bq, 256, 1024, 32, 1e-5f);
  // offset branch -> sampling positions
  offset_conv_kernel<<<64, 256, 0, stream>>>(qn, w_dw, cvb);
  offset_pos_kernel<<<64, 256, 0, stream>>>(cvb, g_off, b_off, w_off, posb);
  // deformable K/V sampling
  sample_kernel<<<64, 256, 0, stream>>>(key, value, posb, ksb, vsb);
  // kh = GN32(conv1x1(k_s, w_k)) ; vh likewise
  gemm_wmma_kernel<0><<<dim3(4, 16, 16), gblk, 0, stream>>>(w_k, ksb, khb, 256, 256);
  gn_kernel<<<16 * 32, 256, 0, stream>>>(khb, gk, bk, 256, 256, 32, 1e-5f);
  gemm_wmma_kernel<0><<<dim3(4, 16, 16), gblk, 0, stream>>>(w_v, vsb, vhb, 256, 256);
  gn_kernel<<<16 * 32, 256, 0, stream>>>(vhb, gv, bv, 256, 256, 32, 1e-5f);
  // fused attention (scores + rpe bias + softmax + PV)
  attn_kernel<<<dim3(64, 128), 256, 0, stream>>>(qn, khb, vhb, posb, rpe, aob);
  // out = GN32(conv1x1(hardswish(ao), w_out))
  gemm_wmma_kernel<1><<<dim3(16, 16, 16), gblk, 0, stream>>>(w_out, aob, out, 256, 1024);
  gn_kernel<<<16 * 32, 256, 0, stream>>>(out, go, bo, 256, 1024, 32, 1e-5f);
}